// mulatt_88433376624678
// MI455X (gfx1250) — compile-verified
//
#include <hip/hip_runtime.h>
#include <hip/hip_bf16.h>
#include <math.h>

// Problem constants (from reference setup_inputs)
#define BDIM 4
#define CDIM 32
#define SDIM 64
#define HDIM 256
#define NEG_BIG (-1e10f)

// LDS row stride: 64 + 4 pad -> bank = (4*row + col) % 64, conflict-free for
// the 16-row / 16-col WMMA fragment access patterns, keeps 16B alignment.
// TDM reproduces this layout in hardware: pad 4 DWORDs every 64 DWORDs.
#define LSTR 68

typedef __attribute__((ext_vector_type(2))) float        v2f;
typedef __attribute__((ext_vector_type(8))) float        v8f;
typedef __attribute__((ext_vector_type(4))) int          v4i;
typedef __attribute__((ext_vector_type(8))) int          v8i;
typedef __attribute__((ext_vector_type(4))) unsigned int v4u;

#if defined(__AMDGCN__) && __has_builtin(__builtin_amdgcn_tensor_load_to_lds)
#define HAVE_TDM 1
#else
#define HAVE_TDM 0
#endif

__device__ __forceinline__ float sigmoidf_fast(float x) {
    return 1.0f / (1.0f + __expf(-x));
}

// ---------------------------------------------------------------------------
// CDNA5 async global<->LDS copies (ASYNCcnt) — used for output epilogues and
// as staging fallback when TDM is unavailable.
// ---------------------------------------------------------------------------
__device__ __forceinline__ void async_copy16(const float* gsrc, float* ldst) {
#if defined(__AMDGCN__)
#if __has_builtin(__builtin_amdgcn_global_load_async_to_lds_b128)
    __builtin_amdgcn_global_load_async_to_lds_b128(
        (__attribute__((address_space(1))) v4i*)gsrc,
        (__attribute__((address_space(3))) v4i*)ldst, 0, 0);
#else
    unsigned l32 = (unsigned)(unsigned long long)(const void*)ldst;
    asm volatile("global_load_async_to_lds_b128 %0, %1, off"
                 :
                 : "v"(l32), "v"((unsigned long long)gsrc)
                 : "memory");
#endif
#else
    (void)gsrc; (void)ldst;
#endif
}

__device__ __forceinline__ void async_store16(float* gdst, const float* lsrc) {
#if defined(__AMDGCN__)
#if __has_builtin(__builtin_amdgcn_global_store_async_from_lds_b128)
    __builtin_amdgcn_global_store_async_from_lds_b128(
        (__attribute__((address_space(1))) v4i*)gdst,
        (__attribute__((address_space(3))) v4i*)lsrc, 0, 0);
#else
    unsigned l32 = (unsigned)(unsigned long long)(const void*)lsrc;
    asm volatile("global_store_async_from_lds_b128 %0, %1, off"
                 :
                 : "v"((unsigned long long)gdst), "v"(l32)
                 : "memory");
#endif
#else
    (void)gdst; (void)lsrc;
#endif
}

__device__ __forceinline__ void wait_async0() {
#if defined(__AMDGCN__)
#if __has_builtin(__builtin_amdgcn_s_wait_asynccnt)
    __builtin_amdgcn_s_wait_asynccnt(0);
#else
    asm volatile("s_wait_asynccnt 0x0" ::: "memory");
#endif
#endif
}

__device__ __forceinline__ void wait_tensor0() {
#if defined(__AMDGCN__)
#if __has_builtin(__builtin_amdgcn_s_wait_tensorcnt)
    __builtin_amdgcn_s_wait_tensorcnt(0);
#else
    asm volatile("s_wait_tensorcnt 0x0" ::: "memory");
#endif
#endif
}

// ---------------------------------------------------------------------------
// TDM: DMA a 64x64 f32 tile (rows stride 256 elems in global) into LDS with
// 4-DWORD padding after every 64 DWORDs -> LDS row stride LSTR=68 floats.
// D# packing per CDNA5 ISA §8 (group0 128b, group1 256b, groups 2/3 zero).
// This toolchain exposes the 6-arg builtin (extra zero descriptor group).
// ---------------------------------------------------------------------------
#if HAVE_TDM
__device__ __forceinline__ void tdm_load_64x64(const float* gtile, const float* ltile) {
    const unsigned long long ga = (unsigned long long)gtile;
    const unsigned lds = (unsigned)(unsigned long long)(const void*)ltile;

    v4u g0;
    g0.x = 1u;                                    // count=1 (valid), user mode
    g0.y = lds;                                   // lds_addr [63:32]
    g0.z = (unsigned)ga;                          // global_addr [95:64]
    g0.w = (unsigned)((ga >> 32) & 0x01FFFFFFull) // global_addr [120:96]
         | (2u << 30);                            // type=2 ("image") [127:126]

    v8i g1;
    g1[0] = (int)((2u << 16)      // data_size = 4 bytes
                | (1u << 20)      // pad_enable
                | (5u << 22)      // pad_interval: 2^(5+1) = 64 DWORDs
                | (3u << 25));    // pad_amount: 4 DWORDs
    g1[1] = (int)(256u << 16);    // tensor_dim0 = 256 (bits 79:48)
    g1[2] = (int)(64u << 16);     // tensor_dim1 = 64  (bits 111:80)
    g1[3] = (int)(64u << 16);     // tile_dim0 = 64    (bits 127:112)
    g1[4] = 64;                   // tile_dim1 = 64, tile_dim2 = 0
    g1[5] = 256;                  // tensor_dim0_stride = 256 (bits 207:160)
    g1[6] = 0;                    // stride hi / tensor_dim1_stride lo
    g1[7] = 0;                    // tensor_dim1_stride hi

    v4i gz4 = {0, 0, 0, 0};                       // groups 2/3: 2-D tensor
    v8i gz8 = {0, 0, 0, 0, 0, 0, 0, 0};           // extra group (unused)
    __builtin_amdgcn_tensor_load_to_lds(g0, g1, gz4, gz4, gz8, 0);
}
#endif

// Fallback staging: per-thread async copies of one 64x64 chunk pair.
__device__ __forceinline__ void stage_chunk_async(const float* ga, float* la,
                                                  const float* gb, float* lb,
                                                  int sr, int sq) {
    const float* gar = ga + (size_t)sr * HDIM;
    const float* gbr = gb + (size_t)sr * HDIM;
    float* lar = la + sr * LSTR;
    float* lbr = lb + sr * LSTR;
#pragma unroll
    for (int u = 0; u < 4; ++u) {
        const int c = (sq + 4 * u) * 4;
        async_copy16(gar + c, lar + c);
        async_copy16(gbr + c, lbr + c);
    }
}

// ---------------------------------------------------------------------------
// Kernel 1: proj[r, g] = sum_h X[r, h] * W[g, h]
//   Grid (128, 4); block 256 threads = 8 waves; 64x64 tile, K in 4 chunks,
//   double-buffered TDM (or async) staging overlapped with WMMA compute.
// ---------------------------------------------------------------------------
__global__ __launch_bounds__(256)
void proj_gemm_kernel(const float* __restrict__ X,
                      const float* __restrict__ W,
                      float* __restrict__ P) {
    __shared__ float Xs[2][64 * LSTR];
    __shared__ float Ws[2][64 * LSTR];

    const int tid  = threadIdx.x;
    const int lane = tid & 31;
    const int wv   = tid >> 5;
    const int hi   = lane >> 4;
    const int lo   = lane & 15;

    const int rowBase = blockIdx.x * 64;
    const int colBase = blockIdx.y * 64;

    const int mt = wv & 3;
    const int n0 = (wv >> 2) * 2;

    v8f acc0 = {0.f,0.f,0.f,0.f,0.f,0.f,0.f,0.f};
    v8f acc1 = {0.f,0.f,0.f,0.f,0.f,0.f,0.f,0.f};

    const int sr = tid >> 2;
    const int sq = tid & 3;

    const float* gX = X + (size_t)rowBase * HDIM;
    const float* gW = W + (size_t)colBase * HDIM;

#if HAVE_TDM
    if (wv == 0) { tdm_load_64x64(gX, Xs[0]); tdm_load_64x64(gW, Ws[0]); }
#else
    stage_chunk_async(gX, Xs[0], gW, Ws[0], sr, sq);
#endif

    for (int kc = 0; kc < 4; ++kc) {
        const int cur = kc & 1;
        const int nxt = cur ^ 1;
#if HAVE_TDM
        if (wv == 0) wait_tensor0();
#else
        wait_async0();
#endif
        __syncthreads();  // chunk kc visible to all; buf 'nxt' free for refill
        if (kc < 3) {
#if HAVE_TDM
            if (wv == 0) {
                tdm_load_64x64(gX + (kc + 1) * 64, Xs[nxt]);
                tdm_load_64x64(gW + (kc + 1) * 64, Ws[nxt]);
            }
#else
            stage_chunk_async(gX + (kc + 1) * 64, Xs[nxt],
                              gW + (kc + 1) * 64, Ws[nxt], sr, sq);
#endif
        }

        const float* Xc = Xs[cur];
        const float* Wc = Ws[cur];
#pragma unroll 4
        for (int k4 = 0; k4 < 64; k4 += 4) {
            const int kk = k4 + 2 * hi;
            const float* ap  = Xc + (mt * 16 + lo) * LSTR + kk;
            const float* b0p = Wc + (n0 * 16 + lo) * LSTR + kk;
            const float* b1p = Wc + ((n0 + 1) * 16 + lo) * LSTR + kk;
            v2f a;  a.x  = ap[0];  a.y  = ap[1];
            v2f b0; b0.x = b0p[0]; b0.y = b0p[1];
            v2f b1; b1.x = b1p[0]; b1.y = b1p[1];
            acc0 = __builtin_amdgcn_wmma_f32_16x16x4_f32(false, a, false, b0,
                                                         (short)0, acc0, false, false);
            acc1 = __builtin_amdgcn_wmma_f32_16x16x4_f32(false, a, false, b1,
                                                         (short)0, acc1, false, false);
        }
    }

    // Stage result tile in LDS (buf 0 is idle), then coalesced async store.
    __syncthreads();
#pragma unroll
    for (int r = 0; r < 8; ++r) {
        const int srow = mt * 16 + r + 8 * hi;
        Xs[0][srow * LSTR + n0 * 16 + lo]      = acc0[r];
        Xs[0][srow * LSTR + n0 * 16 + lo + 16] = acc1[r];
    }
    __syncthreads();
    {
        float* gdst = P + (size_t)(rowBase + sr) * HDIM + colBase + sq * 16;
        const float* lsrc = Xs[0] + sr * LSTR + sq * 16;
#pragma unroll
        for (int u = 0; u < 4; ++u) {
            async_store16(gdst + 4 * u, lsrc + 4 * u);
        }
    }
    // s_endpgm performs an implicit wait-idle, draining ASYNCcnt.
}

// ---------------------------------------------------------------------------
// Kernel 2: one block per (b, i, j).
//   scores[s,t] = sum_h P[b,i,s,h] * X[b,j,t,h]     (64x64x256 WMMA GEMM)
//   att = sigmoid(scores + (1-mask[b,j,t])*NEG_BIG) -> LDS -> async store
//   w[t] = sum_s att_self[b,i,s] * att[s,t]
//   fused[h] = sum_t w[t] * X[b,j,t,h]              -> outFused
// ---------------------------------------------------------------------------
__global__ __launch_bounds__(256)
void att_fused_kernel(const float* __restrict__ X,      // inputs [B,C,S,H]
                      const float* __restrict__ mask,   // [B,C,S]
                      const float* __restrict__ aself,  // [B,C,S]
                      const float* __restrict__ P,      // proj   [B,C,S,H]
                      float* __restrict__ outFused,     // [B,C,C,H]
                      float* __restrict__ outAtt) {     // [B,C,C,S,S]
    __shared__ float Ps[2][64 * LSTR];
    __shared__ float Xs[2][64 * LSTR];
    __shared__ float At[64 * LSTR];
    __shared__ float maskAdd[SDIM];
    __shared__ float selfW[SDIM];
    __shared__ float wvec[SDIM];

    const int j = blockIdx.x;
    const int i = blockIdx.y;
    const int b = blockIdx.z;

    const int tid  = threadIdx.x;
    const int lane = tid & 31;
    const int wv   = tid >> 5;
    const int hi   = lane >> 4;
    const int lo   = lane & 15;
    const int mt   = wv & 3;
    const int n0   = (wv >> 2) * 2;

    const size_t baseI = (size_t)(b * CDIM + i) * SDIM * HDIM;  // proj rows
    const size_t baseJ = (size_t)(b * CDIM + j) * SDIM * HDIM;  // key rows

    if (tid < SDIM) {
        maskAdd[tid] = (1.0f - mask[(b * CDIM + j) * SDIM + tid]) * NEG_BIG;
        selfW[tid]   = aself[(b * CDIM + i) * SDIM + tid];
    }

    v8f acc0 = {0.f,0.f,0.f,0.f,0.f,0.f,0.f,0.f};
    v8f acc1 = {0.f,0.f,0.f,0.f,0.f,0.f,0.f,0.f};

    const int sr = tid >> 2;
    const int sq = tid & 3;

    const float* gP = P + baseI;
    const float* gX = X + baseJ;

#if HAVE_TDM
    if (wv == 0) { tdm_load_64x64(gP, Ps[0]); tdm_load_64x64(gX, Xs[0]); }
#else
    stage_chunk_async(gP, Ps[0], gX, Xs[0], sr, sq);
#endif

    for (int kc = 0; kc < 4; ++kc) {
        const int cur = kc & 1;
        const int nxt = cur ^ 1;
#if HAVE_TDM
        if (wv == 0) wait_tensor0();
#else
        wait_async0();
#endif
        __syncthreads();
        if (kc < 3) {
#if HAVE_TDM
            if (wv == 0) {
                tdm_load_64x64(gP + (kc + 1) * 64, Ps[nxt]);
                tdm_load_64x64(gX + (kc + 1) * 64, Xs[nxt]);
            }
#else
            stage_chunk_async(gP + (kc + 1) * 64, Ps[nxt],
                              gX + (kc + 1) * 64, Xs[nxt], sr, sq);
#endif
        }

        const float* Pc = Ps[cur];
        const float* Xc = Xs[cur];
#pragma unroll 4
        for (int k4 = 0; k4 < 64; k4 += 4) {
            const int kk = k4 + 2 * hi;
            const float* ap  = Pc + (mt * 16 + lo) * LSTR + kk;
            const float* b0p = Xc + (n0 * 16 + lo) * LSTR + kk;
            const float* b1p = Xc + ((n0 + 1) * 16 + lo) * LSTR + kk;
            v2f a;  a.x  = ap[0];  a.y  = ap[1];
            v2f b0; b0.x = b0p[0]; b0.y = b0p[1];
            v2f b1; b1.x = b1p[0]; b1.y = b1p[1];
            acc0 = __builtin_amdgcn_wmma_f32_16x16x4_f32(false, a, false, b0,
                                                         (short)0, acc0, false, false);
            acc1 = __builtin_amdgcn_wmma_f32_16x16x4_f32(false, a, false, b1,
                                                         (short)0, acc1, false, false);
        }
    }

    // Epilogue: mask + sigmoid into the LDS att tile.
#pragma unroll
    for (int r = 0; r < 8; ++r) {
        const int s  = mt * 16 + r + 8 * hi;
        const int t0 = n0 * 16 + lo;
        const int t1 = t0 + 16;
        At[s * LSTR + t0] = sigmoidf_fast(acc0[r] + maskAdd[t0]);
        At[s * LSTR + t1] = sigmoidf_fast(acc1[r] + maskAdd[t1]);
    }
    __syncthreads();

    // att tile: LDS -> global, coalesced async b128 stores.
    const size_t attBase =
        ((size_t)((b * CDIM + i) * CDIM + j)) * (size_t)(SDIM * SDIM);
    {
        float* gdst = outAtt + attBase + (size_t)sr * SDIM + sq * 16;
        const float* lsrc = At + sr * LSTR + sq * 16;
#pragma unroll
        for (int u = 0; u < 4; ++u) {
            async_store16(gdst + 4 * u, lsrc + 4 * u);
        }
    }

    // w[t] = sum_s att_self[b,i,s] * att[s,t]
    if (tid < SDIM) {
        float acc = 0.0f;
#pragma unroll 8
        for (int s = 0; s < SDIM; ++s) {
            acc += selfW[s] * At[s * LSTR + tid];
        }
        wvec[tid] = acc;
    }
    __syncthreads();

    // fused[h] = sum_t w[t] * X[b,j,t,h]  (h = tid, coalesced over threads)
    {
        float f = 0.0f;
        const float* xj = X + baseJ + tid;
#pragma unroll 8
        for (int t = 0; t < SDIM; ++t) {
            f += wvec[t] * xj[(size_t)t * HDIM];
        }
        outFused[((size_t)(b * CDIM + i) * CDIM + j) * HDIM + tid] = f;
    }
    // s_endpgm performs an implicit wait-idle, draining ASYNCcnt.
}

// ---------------------------------------------------------------------------
extern "C" void kernel_launch(void* const* d_in, const int* in_sizes, int n_in,
                              void* d_out, int out_size, void* d_ws, size_t ws_size,
                              hipStream_t stream) {
    const float* X     = (const float*)d_in[0];  // inputs   [B,C,S,H]
    const float* mask  = (const float*)d_in[1];  // mask     [B,C,S]
    const float* aself = (const float*)d_in[2];  // att_self [B,C,S]
    const float* W     = (const float*)d_in[3];  // W        [H,H]

    float* outFused = (float*)d_out;                                     // [B,C,C,H]
    float* outAtt   = (float*)d_out + (size_t)BDIM * CDIM * CDIM * HDIM; // [B,C,C,S,S]

    float* P = (float*)d_ws;  // proj scratch: 8192*256 floats = 8 MB

    // proj = inputs @ W^T
    dim3 g1((BDIM * CDIM * SDIM) / 64, HDIM / 64);  // (128, 4)
    proj_gemm_kernel<<<g1, 256, 0, stream>>>(X, W, P);

    // per-(b,i,j): scores GEMM + sigmoid + att out + fused out
    dim3 g2(CDIM, CDIM, BDIM);  // (32, 32, 4) = 4096 blocks
    att_fused_kernel<<<g2, 256, 0, stream>>>(X, mask, aself, P, outFused, outAtt);
}